// Decoder_3161095930299
// MI455X (gfx1250) — compile-verified
//
#include <hip/hip_runtime.h>
#include <hip/hip_bf16.h>
#include <math.h>

// ---------------------------------------------------------------------------
// Problem constants (from reference): B=64, S=256, H=512, E=256, V=32000, ML=32256
// ---------------------------------------------------------------------------
#define B_   64
#define S_   256
#define H_   512
#define E_   256
#define V_   32000
#define ML_  32256
#define D2H  (2 * H_)          // 1024  (encoder feature dim)
#define XCAT1 (D2H + E_)       // 1280  (context | embedding)
#define X3H  (3 * H_)          // 1536  (attn_reading | select_reading), GRU gate dim

typedef float v2f __attribute__((ext_vector_type(2)));
typedef float v4f __attribute__((ext_vector_type(4)));
typedef float v8f __attribute__((ext_vector_type(8)));

// ---------------------------------------------------------------------------
// Double-buffered K-tile for the fp32 WMMA pipeline.
// One tile = KU k-steps (KU*4 k-values): KU A-fragments + 4*KU B-fragments.
// fp32 A 16x4 layout (ISA 7.12.2): lane L<16 holds A[M=L][K={0,1}],
// lane>=16 holds A[M=L-16][K={2,3}].  B (=W^T) mirrors the C/D convention
// (K = vgpr + 2*half, N = lane%16), so both operands are one aligned b64
// load at row (lane&15), cols kbase + 2*(lane>>4) + {0,1} (koff pre-folded
// into the base pointers below).
// ---------------------------------------------------------------------------
#define KU 4

struct KTile {
  v2f a[KU], b0[KU], b1[KU], b2[KU], b3[KU];
};

__device__ __forceinline__ void load_ktile(
    KTile& t, const float* __restrict__ Ar,
    const float* __restrict__ W0, const float* __restrict__ W1,
    const float* __restrict__ W2, const float* __restrict__ W3, int kbase) {
#pragma unroll
  for (int u = 0; u < KU; ++u) {
    const int kk = kbase + 4 * u;
    t.a[u]  = *(const v2f*)(Ar + kk);
    t.b0[u] = *(const v2f*)(W0 + kk);
    t.b1[u] = *(const v2f*)(W1 + kk);
    t.b2[u] = *(const v2f*)(W2 + kk);
    t.b3[u] = *(const v2f*)(W3 + kk);
  }
}

__device__ __forceinline__ void mma_ktile(
    const KTile& t, v8f& acc0, v8f& acc1, v8f& acc2, v8f& acc3) {
#pragma unroll
  for (int u = 0; u < KU; ++u) {
    acc0 = __builtin_amdgcn_wmma_f32_16x16x4_f32(false, t.a[u], false, t.b0[u], (short)0, acc0, false, false);
    acc1 = __builtin_amdgcn_wmma_f32_16x16x4_f32(false, t.a[u], false, t.b1[u], (short)0, acc1, false, false);
    acc2 = __builtin_amdgcn_wmma_f32_16x16x4_f32(false, t.a[u], false, t.b2[u], (short)0, acc2, false, false);
    acc3 = __builtin_amdgcn_wmma_f32_16x16x4_f32(false, t.a[u], false, t.b3[u], (short)0, acc3, false, false);
  }
}

// Runs the full K loop (K % 32 == 0, K/16 even) with software pipelining:
// loads for tile i+1 are issued before the WMMAs of tile i complete, so the
// matrix pipe is never gated on a single in-flight load.
__device__ __forceinline__ void wmma_k_loop(
    const float* __restrict__ Ar,
    const float* __restrict__ W0, const float* __restrict__ W1,
    const float* __restrict__ W2, const float* __restrict__ W3,
    int K, v8f& acc0, v8f& acc1, v8f& acc2, v8f& acc3) {
  KTile t0, t1;
  load_ktile(t0, Ar, W0, W1, W2, W3, 0);
  int k = 0;
#pragma unroll 1
  for (; k + 32 < K; k += 32) {
    load_ktile(t1, Ar, W0, W1, W2, W3, k + 16);
    mma_ktile(t0, acc0, acc1, acc2, acc3);
    load_ktile(t0, Ar, W0, W1, W2, W3, k + 32);
    mma_ktile(t1, acc0, acc1, acc2, acc3);
  }
  load_ktile(t1, Ar, W0, W1, W2, W3, k + 16);
  mma_ktile(t0, acc0, acc1, acc2, acc3);
  mma_ktile(t1, acc0, acc1, acc2, acc3);
}

// ---------------------------------------------------------------------------
// Generic fp32 WMMA GEMM:  C[m, n] = sum_k A[m,k] * W[n,k] + bias[n]
// A: row-major M x K (lda == K).  W: row-major N x K.  C: row stride = ldc.
// One wave computes a 16(M) x 64(N) strip.  grid = (N/64, M/16), block = 32.
// ---------------------------------------------------------------------------
__global__ __launch_bounds__(32) void wmma_gemm_xwt(
    const float* __restrict__ A, const float* __restrict__ W,
    const float* __restrict__ bias, float* __restrict__ C,
    int K, int ldc) {
  const int lane = threadIdx.x;
  const int n0 = blockIdx.x * 64;
  const int m0 = blockIdx.y * 16;
  const int r = lane & 15;
  const int koff = (lane >> 4) << 1;

  const float* Ar = A + (size_t)(m0 + r) * K + koff;
  const float* W0 = W + (size_t)(n0 + r) * K + koff;
  const float* W1 = W0 + (size_t)16 * K;
  const float* W2 = W0 + (size_t)32 * K;
  const float* W3 = W0 + (size_t)48 * K;

  v8f acc0 = {}, acc1 = {}, acc2 = {}, acc3 = {};
  wmma_k_loop(Ar, W0, W1, W2, W3, K, acc0, acc1, acc2, acc3);

  // C/D layout: lane L, element v  ->  row m0 + v + 8*(L>>4), col n0 + (L&15)
  const int col  = lane & 15;
  const int rowb = m0 + 8 * (lane >> 4);
  const float bv0 = bias[n0 + col];
  const float bv1 = bias[n0 + 16 + col];
  const float bv2 = bias[n0 + 32 + col];
  const float bv3 = bias[n0 + 48 + col];
#pragma unroll
  for (int v = 0; v < 8; ++v) {
    float* cr = C + (size_t)(rowb + v) * ldc + n0 + col;
    cr[0]  = acc0[v] + bv0;
    cr[16] = acc1[v] + bv1;
    cr[32] = acc2[v] + bv2;
    cr[48] = acc3[v] + bv3;
  }
}

// ---------------------------------------------------------------------------
// Fused: score_c[b,s] = sum_h tanh( enc[b,s,:] . W_c[h,:] + b_c[h] ) * state[b,h]
// GEMM M=B*S rows of enc (contiguous, K=1024), N=H=512, fully reduced in-wave.
// One wave handles 16 rows x all 512 cols (8 strips of 64).  Never
// materializes the (B,S,H) feature tensor.  grid = (B*S/16), block = 32.
// ---------------------------------------------------------------------------
__global__ __launch_bounds__(32) void score_c_kernel(
    const float* __restrict__ enc, const float* __restrict__ Wc,
    const float* __restrict__ bc, const float* __restrict__ state,
    float* __restrict__ score_c) {
  const int lane = threadIdx.x;
  const int m0 = blockIdx.x * 16;          // row tile in [0, B*S); never straddles b
  const int b = m0 >> 8;                   // /S_
  const int r = lane & 15;
  const int koff = (lane >> 4) << 1;
  const int K = D2H;

  const float* Ar  = enc + (size_t)(m0 + r) * K + koff;
  const float* stb = state + (size_t)b * H_;

  float rsum[8] = {0.f, 0.f, 0.f, 0.f, 0.f, 0.f, 0.f, 0.f};
  const int col = lane & 15;

#pragma unroll 1
  for (int n0 = 0; n0 < H_; n0 += 64) {
    const float* W0 = Wc + (size_t)(n0 + r) * K + koff;
    const float* W1 = W0 + (size_t)16 * K;
    const float* W2 = W0 + (size_t)32 * K;
    const float* W3 = W0 + (size_t)48 * K;
    v8f acc0 = {}, acc1 = {}, acc2 = {}, acc3 = {};
    wmma_k_loop(Ar, W0, W1, W2, W3, K, acc0, acc1, acc2, acc3);

    const float s0 = stb[n0 + col],      c0 = bc[n0 + col];
    const float s1 = stb[n0 + 16 + col], c1 = bc[n0 + 16 + col];
    const float s2 = stb[n0 + 32 + col], c2 = bc[n0 + 32 + col];
    const float s3 = stb[n0 + 48 + col], c3 = bc[n0 + 48 + col];
#pragma unroll
    for (int v = 0; v < 8; ++v) {
      rsum[v] += tanhf(acc0[v] + c0) * s0;
      rsum[v] += tanhf(acc1[v] + c1) * s1;
      rsum[v] += tanhf(acc2[v] + c2) * s2;
      rsum[v] += tanhf(acc3[v] + c3) * s3;
    }
  }
  // Reduce each row-sum across the 16 lanes of each half (xor stays in-half).
#pragma unroll
  for (int v = 0; v < 8; ++v) {
    rsum[v] += __shfl_xor(rsum[v], 1, 32);
    rsum[v] += __shfl_xor(rsum[v], 2, 32);
    rsum[v] += __shfl_xor(rsum[v], 4, 32);
    rsum[v] += __shfl_xor(rsum[v], 8, 32);
  }
  const int half = lane >> 4;
  if (col < 8) {
    float outv = rsum[0];
    if (col == 1) outv = rsum[1];
    if (col == 2) outv = rsum[2];
    if (col == 3) outv = rsum[3];
    if (col == 4) outv = rsum[4];
    if (col == 5) outv = rsum[5];
    if (col == 6) outv = rsum[6];
    if (col == 7) outv = rsum[7];
    score_c[(size_t)b * S_ + (m0 & (S_ - 1)) + 8 * half + col] = outv;
  }
}

// ---------------------------------------------------------------------------
// Embedding gather: xcat1[b, 1024:1280] = embed_table[where(idx>=V, 2, idx)]
// ---------------------------------------------------------------------------
__global__ __launch_bounds__(256) void embed_kernel(
    const int* __restrict__ input_idx, const float* __restrict__ embed_table,
    float* __restrict__ xcat1) {
  const int b = blockIdx.x, t = threadIdx.x;  // t in [0, E_)
  int idx = input_idx[b];
  if (idx >= V_) idx = 2;
  xcat1[(size_t)b * XCAT1 + D2H + t] = embed_table[(size_t)idx * E_ + t];
}

// ---------------------------------------------------------------------------
// attn_scores[b,s] = enc[b,s,:] . attn_strength[b,:]   (one wave per (b,s))
// ---------------------------------------------------------------------------
__global__ __launch_bounds__(256) void attn_scores_kernel(
    const float* __restrict__ enc, const float* __restrict__ astr,
    float* __restrict__ scores) {
  const int wave = threadIdx.x >> 5;
  const int lane = threadIdx.x & 31;
  const int gid = blockIdx.x * 8 + wave;     // b*S + s
  const int b = gid >> 8;
  const float* e = enc + (size_t)gid * D2H;
  const float* a = astr + (size_t)b * D2H;
  float sum = 0.f;
  for (int q = lane; q < D2H / 4; q += 32) {
    v4f ev = *(const v4f*)(e + 4 * q);
    v4f av = *(const v4f*)(a + 4 * q);
    sum += ev[0] * av[0] + ev[1] * av[1] + ev[2] * av[2] + ev[3] * av[3];
  }
  sum += __shfl_xor(sum, 16, 32);
  sum += __shfl_xor(sum, 8, 32);
  sum += __shfl_xor(sum, 4, 32);
  sum += __shfl_xor(sum, 2, 32);
  sum += __shfl_xor(sum, 1, 32);
  if (lane == 0) scores[gid] = sum;
}

// ---------------------------------------------------------------------------
// Per-b softmax over S (in place) + select_weights.
// ---------------------------------------------------------------------------
__global__ __launch_bounds__(256) void softmax_s_kernel(
    float* __restrict__ scores, float* __restrict__ selw,
    const int* __restrict__ seq_idx, const int* __restrict__ input_idx,
    const float* __restrict__ pre_prob_c) {
  __shared__ float red[256];
  const int b = blockIdx.x, t = threadIdx.x;
  const float v = scores[(size_t)b * S_ + t];
  red[t] = v; __syncthreads();
  for (int s = 128; s > 0; s >>= 1) {
    if (t < s) red[t] = fmaxf(red[t], red[t + s]);
    __syncthreads();
  }
  const float mx = red[0]; __syncthreads();
  const float e = expf(v - mx);
  red[t] = e; __syncthreads();
  for (int s = 128; s > 0; s >>= 1) {
    if (t < s) red[t] += red[t + s];
    __syncthreads();
  }
  const float inv = 1.f / red[0];
  scores[(size_t)b * S_ + t] = e * inv;
  selw[(size_t)b * S_ + t] =
      (seq_idx[(size_t)b * S_ + t] == input_idx[b]) ? pre_prob_c[(size_t)b * S_ + t] : 0.f;
}

// ---------------------------------------------------------------------------
// context[b,d] = sum_s aw[b,s]*enc[b,s,d] ; select_reading[b,d] likewise.
// context -> xcat1[:, 0:1024],  select_reading -> x[:, 512:1536].
// ---------------------------------------------------------------------------
__global__ __launch_bounds__(256) void weighted_sum_kernel(
    const float* __restrict__ enc, const float* __restrict__ aw,
    const float* __restrict__ sw, float* __restrict__ xcat1,
    float* __restrict__ x) {
  __shared__ float w1[S_], w2[S_];
  const int b = blockIdx.x;
  const int d = blockIdx.y * 256 + threadIdx.x;
  w1[threadIdx.x] = aw[(size_t)b * S_ + threadIdx.x];
  w2[threadIdx.x] = sw[(size_t)b * S_ + threadIdx.x];
  __syncthreads();
  const float* e = enc + (size_t)b * S_ * D2H + d;
  float c = 0.f, s = 0.f;
#pragma unroll 8
  for (int i = 0; i < S_; ++i) {
    const float ev = e[(size_t)i * D2H];
    c += w1[i] * ev;
    s += w2[i] * ev;
  }
  xcat1[(size_t)b * XCAT1 + d] = c;
  x[(size_t)b * X3H + H_ + d] = s;
}

// ---------------------------------------------------------------------------
// GRU gates -> state (written into d_out's state region).
// ---------------------------------------------------------------------------
__global__ __launch_bounds__(256) void gru_gate_kernel(
    const float* __restrict__ gi, const float* __restrict__ gh,
    const float* __restrict__ pre_state, float* __restrict__ state) {
  const int b = blockIdx.x;
  const int h = blockIdx.y * 256 + threadIdx.x;
  const float* gib = gi + (size_t)b * X3H;
  const float* ghb = gh + (size_t)b * X3H;
  const float r = 1.f / (1.f + expf(-(gib[h] + ghb[h])));
  const float z = 1.f / (1.f + expf(-(gib[H_ + h] + ghb[H_ + h])));
  const float n = tanhf(gib[2 * H_ + h] + r * ghb[2 * H_ + h]);
  state[(size_t)b * H_ + h] = (1.f - z) * n + z * pre_state[(size_t)b * H_ + h];
}

// ---------------------------------------------------------------------------
// Per-b softmax over [score_g | score_c] (32256 values): mx/den, plus the
// copy-distribution  prob_c[b,j] = sum_i prob_c0[b,i]*(idx[i]==idx[j]).
// ---------------------------------------------------------------------------
__global__ __launch_bounds__(256) void softmax_full_kernel(
    const float* __restrict__ score_g, const float* __restrict__ score_c,
    const int* __restrict__ seq_idx, float* __restrict__ mxden,
    float* __restrict__ prob_c_out) {
  __shared__ float red[256];
  __shared__ float pc0[S_];
  __shared__ int sidx[S_];
  const int b = blockIdx.x, t = threadIdx.x;
  const float* sg = score_g + (size_t)b * V_;
  const float* sc = score_c + (size_t)b * S_;
  float mx = -INFINITY;
  for (int i = t; i < V_; i += 256) mx = fmaxf(mx, sg[i]);
  mx = fmaxf(mx, sc[t]);
  red[t] = mx; __syncthreads();
  for (int s = 128; s > 0; s >>= 1) {
    if (t < s) red[t] = fmaxf(red[t], red[t + s]);
    __syncthreads();
  }
  mx = red[0]; __syncthreads();
  float sum = 0.f;
  for (int i = t; i < V_; i += 256) sum += expf(sg[i] - mx);
  const float ec = expf(sc[t] - mx);
  sum += ec;
  red[t] = sum; __syncthreads();
  for (int s = 128; s > 0; s >>= 1) {
    if (t < s) red[t] += red[t + s];
    __syncthreads();
  }
  const float den = red[0];
  if (t == 0) { mxden[b * 2] = mx; mxden[b * 2 + 1] = den; }
  const float inv = 1.f / den;
  pc0[t] = ec * inv;
  sidx[t] = seq_idx[(size_t)b * S_ + t];
  __syncthreads();
  const int my = sidx[t];
  float acc = 0.f;
  for (int i = 0; i < S_; ++i) acc += (sidx[i] == my) ? pc0[i] : 0.f;
  prob_c_out[(size_t)b * S_ + t] = acc;
}

// ---------------------------------------------------------------------------
// out[b, c] = (c < V) ? prob_g[b,c] : 0   for all c in [0, ML)
// ---------------------------------------------------------------------------
__global__ __launch_bounds__(256) void out_fill_kernel(
    const float* __restrict__ score_g, const float* __restrict__ mxden,
    float* __restrict__ out) {
  const int b = blockIdx.x;
  const int c = blockIdx.y * 256 + threadIdx.x;
  const float mx = mxden[b * 2];
  const float inv = 1.f / mxden[b * 2 + 1];
  float v = 0.f;
  if (c < V_) v = expf(score_g[(size_t)b * V_ + c] - mx) * inv;
  out[(size_t)b * ML_ + c] = v;
}

// ---------------------------------------------------------------------------
// Scatter (set-then-add semantics): out[b, idx[b,j]] = prob_c[b,j] + prob_g[idx]
// Duplicate indices write identical values -> benign.
// ---------------------------------------------------------------------------
__global__ __launch_bounds__(256) void scatter_kernel(
    const int* __restrict__ seq_idx, const float* __restrict__ prob_c,
    const float* __restrict__ score_g, const float* __restrict__ mxden,
    float* __restrict__ out) {
  const int b = blockIdx.x, j = threadIdx.x;
  const int c = seq_idx[(size_t)b * S_ + j];
  const float mx = mxden[b * 2];
  const float inv = 1.f / mxden[b * 2 + 1];
  float pg = 0.f;
  if (c < V_) pg = expf(score_g[(size_t)b * V_ + c] - mx) * inv;
  out[(size_t)b * ML_ + c] = prob_c[(size_t)b * S_ + j] + pg;
}

// ---------------------------------------------------------------------------
extern "C" void kernel_launch(void* const* d_in, const int* in_sizes, int n_in,
                              void* d_out, int out_size, void* d_ws, size_t ws_size,
                              hipStream_t stream) {
  (void)in_sizes; (void)n_in; (void)out_size; (void)ws_size;
  const int*   input_idx   = (const int*)d_in[0];
  const float* enc         = (const float*)d_in[1];
  const int*   seq_idx     = (const int*)d_in[2];
  const float* pre_state   = (const float*)d_in[3];
  const float* pre_prob_c  = (const float*)d_in[4];
  const float* embed_table = (const float*)d_in[5];
  const float* W_aw = (const float*)d_in[6];
  const float* b_aw = (const float*)d_in[7];
  const float* W_ac = (const float*)d_in[8];
  const float* b_ac = (const float*)d_in[9];
  const float* W_ih = (const float*)d_in[10];
  const float* b_ih = (const float*)d_in[11];
  const float* W_hh = (const float*)d_in[12];
  const float* b_hh = (const float*)d_in[13];
  const float* W_o  = (const float*)d_in[14];
  const float* b_o  = (const float*)d_in[15];
  const float* W_c  = (const float*)d_in[16];
  const float* b_c  = (const float*)d_in[17];

  // Output regions
  float* out        = (float*)d_out;                         // (B, ML)
  float* state      = out + (size_t)B_ * ML_;                // (B, H)
  float* prob_c_out = state + (size_t)B_ * H_;               // (B, S)

  // Workspace layout (floats)
  float* ws = (float*)d_ws;
  float* astr    = ws;                         // B x 1024
  float* scores  = astr   + (size_t)B_ * D2H;  // B x 256  (-> attn weights in place)
  float* selw    = scores + (size_t)B_ * S_;   // B x 256
  float* xcat1   = selw   + (size_t)B_ * S_;   // B x 1280
  float* x       = xcat1  + (size_t)B_ * XCAT1;// B x 1536
  float* gi      = x      + (size_t)B_ * X3H;  // B x 1536
  float* gh      = gi     + (size_t)B_ * X3H;  // B x 1536
  float* score_g = gh     + (size_t)B_ * X3H;  // B x 32000
  float* score_c = score_g+ (size_t)B_ * V_;   // B x 256
  float* mxden   = score_c+ (size_t)B_ * S_;   // B x 2

  // 1) embedding -> xcat1[:, 1024:1280]
  embed_kernel<<<dim3(B_), dim3(E_), 0, stream>>>(input_idx, embed_table, xcat1);
  // 2) attn_strength = pre_state @ W_aw.T + b_aw        (64 x 1024, K=512)
  wmma_gemm_xwt<<<dim3(D2H / 64, B_ / 16), dim3(32), 0, stream>>>(
      pre_state, W_aw, b_aw, astr, H_, D2H);
  // 3) attn_scores
  attn_scores_kernel<<<dim3(B_ * S_ / 8), dim3(256), 0, stream>>>(enc, astr, scores);
  // 4) softmax over S + select weights
  softmax_s_kernel<<<dim3(B_), dim3(S_), 0, stream>>>(scores, selw, seq_idx, input_idx, pre_prob_c);
  // 5) context -> xcat1[:,0:1024], select_reading -> x[:,512:1536]
  weighted_sum_kernel<<<dim3(B_, D2H / 256), dim3(256), 0, stream>>>(enc, scores, selw, xcat1, x);
  // 6) attn_reading = xcat1 @ W_ac.T + b_ac -> x[:, 0:512]   (64 x 512, K=1280)
  wmma_gemm_xwt<<<dim3(H_ / 64, B_ / 16), dim3(32), 0, stream>>>(
      xcat1, W_ac, b_ac, x, XCAT1, X3H);
  // 7) gi = x @ W_ih.T + b_ih                        (64 x 1536, K=1536)
  wmma_gemm_xwt<<<dim3(X3H / 64, B_ / 16), dim3(32), 0, stream>>>(
      x, W_ih, b_ih, gi, X3H, X3H);
  // 8) gh = pre_state @ W_hh.T + b_hh                (64 x 1536, K=512)
  wmma_gemm_xwt<<<dim3(X3H / 64, B_ / 16), dim3(32), 0, stream>>>(
      pre_state, W_hh, b_hh, gh, H_, X3H);
  // 9) GRU gates -> state (in d_out)
  gru_gate_kernel<<<dim3(B_, H_ / 256), dim3(256), 0, stream>>>(gi, gh, pre_state, state);
  // 10) score_g = state @ W_o.T + b_o                (64 x 32000, K=512)
  wmma_gemm_xwt<<<dim3(V_ / 64, B_ / 16), dim3(32), 0, stream>>>(
      state, W_o, b_o, score_g, H_, V_);
  // 11) fused tanh(enc @ W_c.T + b_c) . state -> score_c   (16384x512, K=1024)
  score_c_kernel<<<dim3(B_ * S_ / 16), dim3(32), 0, stream>>>(enc, W_c, b_c, state, score_c);
  // 12) full softmax stats + copy-distribution prob_c (into d_out)
  softmax_full_kernel<<<dim3(B_), dim3(256), 0, stream>>>(
      score_g, score_c, seq_idx, mxden, prob_c_out);
  // 13) fill out with prob_g (zeros beyond V)
  out_fill_kernel<<<dim3(B_, ML_ / 256), dim3(256), 0, stream>>>(score_g, mxden, out);
  // 14) scatter prob_c (+ prob_g at those positions), set semantics
  scatter_kernel<<<dim3(B_), dim3(S_), 0, stream>>>(seq_idx, prob_c_out, score_g, mxden, out);
}